// GNNStack_31842887533162
// MI455X (gfx1250) — compile-verified
//
#include <hip/hip_runtime.h>
#include <hip/hip_bf16.h>
#include <math.h>

typedef __attribute__((ext_vector_type(2))) float v2f;
typedef __attribute__((ext_vector_type(8))) float v8f;

#define C_DIM 128
#define TILE_ROWS 16
#define TILES_PER_WAVE 2                       // 32 rows per wave
#define WAVES_PER_BLOCK 4                      // 128 threads
#define ROWS_PER_BLOCK (TILE_ROWS * TILES_PER_WAVE * WAVES_PER_BLOCK)  // 128

// ---------------------------------------------------------------------------
// Kernel 1: inv_sqrt_deg[n] = rsqrt(1 + #valid neighbors)
// ---------------------------------------------------------------------------
__global__ __launch_bounds__(256) void gcn_deg_kernel(
    const int* __restrict__ edges, float* __restrict__ inv_s, int N) {
  int n = blockIdx.x * blockDim.x + threadIdx.x;
  if (n >= N) return;
  int cnt = 1;
#pragma unroll
  for (int d = 0; d < 16; ++d) cnt += (edges[n * 16 + d] >= 0) ? 1 : 0;
  inv_s[n] = rsqrtf((float)cnt);
}

// ---------------------------------------------------------------------------
// Kernel 2: H = (X @ W + b) * inv_sqrt_deg  via V_WMMA_F32_16X16X4_F32
// W staged in LDS, K-pair interleaved so each B fragment is one ds_load_b64:
//   lWp[(k/2)*128 + col] = { W[k][col], W[k+1][col] }   (k even)
// Each wave computes a 32x128 stripe (two 16-row A tiles sharing B frags).
// ---------------------------------------------------------------------------
__global__ __launch_bounds__(128) void gcn_gemm_wmma_kernel(
    const float* __restrict__ X, const float* __restrict__ W,
    const float* __restrict__ bias, const float* __restrict__ inv_s,
    float* __restrict__ H, int N) {
  __shared__ v2f lWp[(C_DIM / 2) * C_DIM];  // 64 KB, K-pair interleaved

  const int tid = threadIdx.x;

  // Cooperative repack of W into LDS: coalesced row-pair reads, b64 LDS writes.
  for (int idx = tid; idx < (C_DIM / 2) * C_DIM; idx += 128) {
    const int k2 = idx >> 7;         // K pair index
    const int col = idx & (C_DIM - 1);
    v2f p;
    p.x = W[(2 * k2 + 0) * C_DIM + col];
    p.y = W[(2 * k2 + 1) * C_DIM + col];
    lWp[idx] = p;
  }
  __syncthreads();

  const int wave = tid >> 5;
  const int lane = tid & 31;
  const int mrow = lane & 15;   // M index (A) / N index (B,C)
  const int khalf = lane >> 4;  // 0: K={0,1}, 1: K={2,3} within a k-step

  const int row0 = blockIdx.x * ROWS_PER_BLOCK + wave * (TILE_ROWS * TILES_PER_WAVE);
  int arowA = row0 + mrow;
  int arowB = row0 + TILE_ROWS + mrow;
  if (arowA >= N) arowA = 0;  // clamp loads; stores predicated later
  if (arowB >= N) arowB = 0;
  const float* xrowA = X + (size_t)arowA * C_DIM;
  const float* xrowB = X + (size_t)arowB * C_DIM;

  v8f acc0[8], acc1[8];
  const v8f vzero = {0.f, 0.f, 0.f, 0.f, 0.f, 0.f, 0.f, 0.f};
#pragma unroll
  for (int t = 0; t < 8; ++t) {
    acc0[t] = vzero;
    acc1[t] = vzero;
  }

#pragma unroll 2
  for (int k0 = 0; k0 < C_DIM; k0 += 4) {
    // A fragments (16x4 f32): lane holds X[row][k0 + 2*khalf + {0,1}] — one b64.
    const v2f afragA = *(const v2f*)(xrowA + k0 + 2 * khalf);
    const v2f afragB = *(const v2f*)(xrowB + k0 + 2 * khalf);
    const v2f* bbase = &lWp[(k0 / 2 + khalf) * C_DIM];
#pragma unroll
    for (int t = 0; t < 8; ++t) {
      // B fragment (4x16 f32): VGPR v -> K = k0 + v + 2*khalf; one ds_load_b64.
      const v2f bfrag = bbase[t * 16 + mrow];
      acc0[t] = __builtin_amdgcn_wmma_f32_16x16x4_f32(
          false, afragA, false, bfrag, (short)0, acc0[t], false, false);
      acc1[t] = __builtin_amdgcn_wmma_f32_16x16x4_f32(
          false, afragB, false, bfrag, (short)0, acc1[t], false, false);
    }
  }

  // Epilogue: element (vgpr v, lane) -> row = rowbase + v + 8*khalf, col = t*16 + mrow
#pragma unroll
  for (int tile = 0; tile < TILES_PER_WAVE; ++tile) {
    const v8f* acc = (tile == 0) ? acc0 : acc1;
#pragma unroll
    for (int v = 0; v < 8; ++v) {
      const int r = row0 + tile * TILE_ROWS + 8 * khalf + v;
      if (r < N) {
        const float sc = inv_s[r];
        float* hrow = H + (size_t)r * C_DIM;
#pragma unroll
        for (int t = 0; t < 8; ++t) {
          const int col = t * 16 + mrow;
          hrow[col] = (acc[t][v] + bias[col]) * sc;
        }
      }
    }
  }
}

// ---------------------------------------------------------------------------
// Kernel 3: out[n] = ELU((sum_{d: e>=0} H[e[n,d]] + H[n]) * inv_sqrt_deg[n])
// 32 lanes per node, one float4 channel-chunk per lane (512B coalesced rows,
// L2-resident since H is 51 MB < 192 MB L2).
// ---------------------------------------------------------------------------
__global__ __launch_bounds__(256) void gcn_gather_elu_kernel(
    const float* __restrict__ H, const int* __restrict__ edges,
    const float* __restrict__ inv_s, float* __restrict__ out, int N) {
  const int node = blockIdx.x * 8 + (threadIdx.x >> 5);
  if (node >= N) return;
  const int c4 = threadIdx.x & 31;  // float4 chunk: cols 4*c4 .. 4*c4+3

  const float4* Hv = (const float4*)H;
  float4 self = Hv[(size_t)node * 32 + c4];
  float ax = self.x, ay = self.y, az = self.z, aw = self.w;

#pragma unroll
  for (int d = 0; d < 16; ++d) {
    const int e = edges[node * 16 + d];
    if (e >= 0) {
      float4 v = Hv[(size_t)e * 32 + c4];
      ax += v.x;
      ay += v.y;
      az += v.z;
      aw += v.w;
    }
  }

  const float sc = inv_s[node];
  ax *= sc;
  ay *= sc;
  az *= sc;
  aw *= sc;

  float4 r;
  r.x = (ax > 0.f) ? ax : expm1f(ax);
  r.y = (ay > 0.f) ? ay : expm1f(ay);
  r.z = (az > 0.f) ? az : expm1f(az);
  r.w = (aw > 0.f) ? aw : expm1f(aw);
  ((float4*)out)[(size_t)node * 32 + c4] = r;
}

// ---------------------------------------------------------------------------
// Launch: deg -> GEMM1 -> gather+ELU(->out) -> GEMM2(reads out) -> gather+ELU
// Workspace: [inv_s: N floats (padded)] [H: N*128 floats]
// ---------------------------------------------------------------------------
extern "C" void kernel_launch(void* const* d_in, const int* in_sizes, int n_in,
                              void* d_out, int out_size, void* d_ws,
                              size_t ws_size, hipStream_t stream) {
  const float* x = (const float*)d_in[0];
  const int* edges = (const int*)d_in[1];
  const float* W1 = (const float*)d_in[2];
  const float* b1 = (const float*)d_in[3];
  const float* W2 = (const float*)d_in[4];
  const float* b2 = (const float*)d_in[5];
  float* out = (float*)d_out;

  const int N = in_sizes[0] / C_DIM;

  float* inv_s = (float*)d_ws;
  const size_t inv_pad = ((size_t)N + 63) & ~(size_t)63;
  float* H = inv_s + inv_pad;

  const int deg_blocks = (N + 255) / 256;
  const int gemm_blocks = (N + ROWS_PER_BLOCK - 1) / ROWS_PER_BLOCK;
  const int gather_blocks = (N + 7) / 8;

  gcn_deg_kernel<<<deg_blocks, 256, 0, stream>>>(edges, inv_s, N);

  // Layer 1
  gcn_gemm_wmma_kernel<<<gemm_blocks, 128, 0, stream>>>(x, W1, b1, inv_s, H, N);
  gcn_gather_elu_kernel<<<gather_blocks, 256, 0, stream>>>(H, edges, inv_s, out, N);

  // Layer 2 (reads layer-1 activations from d_out)
  gcn_gemm_wmma_kernel<<<gemm_blocks, 128, 0, stream>>>(out, W2, b2, inv_s, H, N);
  gcn_gather_elu_kernel<<<gather_blocks, 256, 0, stream>>>(H, edges, inv_s, out, N);
}